// DiffExplainer_1400159339039
// MI455X (gfx1250) — compile-verified
//
#include <hip/hip_runtime.h>
#include <math.h>

typedef float v2f_t __attribute__((ext_vector_type(2)));
typedef float v8f_t __attribute__((ext_vector_type(8)));

#define NNODES 40000
#define NEDGES 640000
#define DIM 128
#define ROWTILES (NNODES / 16)   // 2500
#define WPAD 132                 // padded LDS row stride: bank = (4n + k) % 64 -> conflict-free frags

// ---------------------------------------------------------------------------
// WMMA fp32 GEMM with LDS-staged weights + 8x A-fragment reuse.
// out[n,d] = (optional relu)(A[n,:] . W[d, off:off+128] + bias[d]) (+ resid[n,d])
// A: N x 128 row-major. W element: W[d*w_stride + w_off + k].
// Block = 256 threads = 8 waves; W (128x128) staged once per block in LDS.
// Each wave computes a 16x128 row strip = 8 column tiles, reusing its A
// fragment across all 8 tiles (8 independent WMMA accumulation chains).
// Grid.x = ceil(ROWTILES/8).
// ---------------------------------------------------------------------------
__global__ __launch_bounds__(256)
void gemm_wmma_f32(const float* __restrict__ A,
                   const float* __restrict__ W,
                   int w_stride, int w_off,
                   const float* __restrict__ bias,
                   const float* __restrict__ resid,
                   float* __restrict__ out,
                   int do_relu)
{
    __shared__ float Wl[DIM * WPAD];

    const int wave  = threadIdx.x >> 5;
    const int lane  = threadIdx.x & 31;
    const int l15   = lane & 15;
    const int khalf = (lane >> 4) * 2;       // 0 or 2

    // Cooperatively stage W (128 x 128) into LDS with padded row stride.
    for (int i = threadIdx.x; i < DIM * (DIM / 4); i += 256) {
        const int n  = i >> 5;               // W row (= output column)
        const int k4 = (i & 31) * 4;
        const float4 w = *(const float4*)(W + (size_t)n * w_stride + w_off + k4);
        float* p = &Wl[n * WPAD + k4];
        p[0] = w.x; p[1] = w.y; p[2] = w.z; p[3] = w.w;
    }
    __syncthreads();

    const int tile = blockIdx.x * 8 + wave;  // row tile handled by this wave
    if (tile >= ROWTILES) return;            // wave-uniform; after the barrier
    const int row0 = tile * 16;

    const float* Arow = A + (size_t)(row0 + l15) * DIM;

    v8f_t zero = {};
    v8f_t acc[8];
#pragma unroll
    for (int t = 0; t < 8; ++t) acc[t] = zero;

    for (int k0 = 0; k0 < DIM; k0 += 4) {
        const int kk = k0 + khalf;
        const v2f_t a = *(const v2f_t*)(Arow + kk);   // A[M=l15][kk], [kk+1]
#pragma unroll
        for (int t = 0; t < 8; ++t) {
            const v2f_t b = *(const v2f_t*)(&Wl[(t * 16 + l15) * WPAD + kk]);
            acc[t] = __builtin_amdgcn_wmma_f32_16x16x4_f32(
                         false, a, false, b, (short)0, acc[t], false, false);
        }
    }

    const int rbase = row0 + ((lane >> 4) * 8);
#pragma unroll
    for (int t = 0; t < 8; ++t) {
        const int colN = t * 16 + l15;
        const float bv = bias ? bias[colN] : 0.0f;
#pragma unroll
        for (int r = 0; r < 8; ++r) {
            float v = acc[t][r] + bv;
            if (do_relu) v = v > 0.0f ? v : 0.0f;
            const size_t idx = (size_t)(rbase + r) * DIM + colN;
            if (resid) v += resid[idx];
            out[idx] = v;
        }
    }
}

// hs[n] = h[n,:].ws ; hd[n] = h[n,:].wd  (ws = attw[0:128], wd = attw[128:256])
__global__ void gemv2_kernel(const float* __restrict__ h, const float* __restrict__ attw,
                             float* __restrict__ hs, float* __restrict__ hd, int n)
{
    int i = blockIdx.x * blockDim.x + threadIdx.x;
    if (i >= n) return;
    const float* hr = h + (size_t)i * DIM;
    float s0 = 0.f, s1 = 0.f;
#pragma unroll 8
    for (int k = 0; k < DIM; ++k) {
        float v = hr[k];
        s0 += v * attw[k];
        s1 += v * attw[DIM + k];
    }
    hs[i] = s0;
    hd[i] = s1;
}

// scores + per-block max (grid-stride; every block writes its pmax slot)
__global__ void edge_score_max(const float* __restrict__ hs, const float* __restrict__ hd,
                               const int* __restrict__ src, const int* __restrict__ dst,
                               const float* __restrict__ att_b, int l,
                               float* __restrict__ sc, float* __restrict__ pmax, int E)
{
    __shared__ float sm[256];
    const float bb = att_b[l];
    float m = -3.4e38f;
    for (int e = blockIdx.x * blockDim.x + threadIdx.x; e < E; e += gridDim.x * blockDim.x) {
        float s = hs[src[e]] + hd[dst[e]] + bb;
        s = s > 0.0f ? s : 0.2f * s;       // leaky_relu(0.2)
        sc[e] = s;
        m = fmaxf(m, s);
    }
    sm[threadIdx.x] = m;
    __syncthreads();
    for (int o = 128; o > 0; o >>= 1) {
        if (threadIdx.x < o) sm[threadIdx.x] = fmaxf(sm[threadIdx.x], sm[threadIdx.x + o]);
        __syncthreads();
    }
    if (threadIdx.x == 0) pmax[blockIdx.x] = sm[0];
}

__global__ void reduce_max_k(const float* __restrict__ p, float* __restrict__ outv, int n)
{
    __shared__ float sm[256];
    float m = -3.4e38f;
    for (int i = threadIdx.x; i < n; i += 256) m = fmaxf(m, p[i]);
    sm[threadIdx.x] = m;
    __syncthreads();
    for (int o = 128; o > 0; o >>= 1) {
        if (threadIdx.x < o) sm[threadIdx.x] = fmaxf(sm[threadIdx.x], sm[threadIdx.x + o]);
        __syncthreads();
    }
    if (threadIdx.x == 0) outv[0] = sm[0];
}

__global__ void exp_sum_k(float* __restrict__ sc, const float* __restrict__ mx,
                          float* __restrict__ psum, int E)
{
    __shared__ float sm[256];
    const float m = mx[0];
    float s = 0.f;
    for (int e = blockIdx.x * blockDim.x + threadIdx.x; e < E; e += gridDim.x * blockDim.x) {
        float v = expf(sc[e] - m);
        sc[e] = v;
        s += v;
    }
    sm[threadIdx.x] = s;
    __syncthreads();
    for (int o = 128; o > 0; o >>= 1) {
        if (threadIdx.x < o) sm[threadIdx.x] += sm[threadIdx.x + o];
        __syncthreads();
    }
    if (threadIdx.x == 0) psum[blockIdx.x] = sm[0];
}

__global__ void reduce_sum_k(const float* __restrict__ p, float* __restrict__ outv, int n)
{
    __shared__ float sm[256];
    float s = 0.f;
    for (int i = threadIdx.x; i < n; i += 256) s += p[i];
    sm[threadIdx.x] = s;
    __syncthreads();
    for (int o = 128; o > 0; o >>= 1) {
        if (threadIdx.x < o) sm[threadIdx.x] += sm[threadIdx.x + o];
        __syncthreads();
    }
    if (threadIdx.x == 0) outv[0] = sm[0];
}

__global__ void normalize_k(const float* __restrict__ sc, const float* __restrict__ sum,
                            float* __restrict__ att, int E)
{
    int e = blockIdx.x * blockDim.x + threadIdx.x;
    if (e >= E) return;
    att[e] = sc[e] / sum[0];
}

__global__ void zero_f32(float* __restrict__ p, long long n)
{
    long long i = (long long)blockIdx.x * blockDim.x + threadIdx.x;
    if (i < n) p[i] = 0.0f;
}

// agg[dst] += h[src] * a   (32 lanes per edge, 4 cols each)
__global__ void aggregate_k(const float* __restrict__ h, const int* __restrict__ src,
                            const int* __restrict__ dst, const float* __restrict__ att,
                            float* __restrict__ agg, int E)
{
    int t = blockIdx.x * blockDim.x + threadIdx.x;
    int e = t >> 5;
    if (e >= E) return;
    int c = (t & 31) * 4;
    float a = att[e];
    int s = src[e], d = dst[e];
    const float4 hv = *(const float4*)(h + (size_t)s * DIM + c);
    float* ap = agg + (size_t)d * DIM + c;
    atomicAdd(ap + 0, hv.x * a);
    atomicAdd(ap + 1, hv.y * a);
    atomicAdd(ap + 2, hv.z * a);
    atomicAdd(ap + 3, hv.w * a);
}

__global__ void colsum_k(const float* __restrict__ h, float* __restrict__ csum, int n)
{
    int c = threadIdx.x; // 128 threads
    float s = 0.f;
    for (int r = blockIdx.x; r < n; r += gridDim.x) s += h[(size_t)r * DIM + c];
    atomicAdd(&csum[c], s);
}

__global__ void readout_k(const float* __restrict__ csum,
                          const float* __restrict__ roW1, const float* __restrict__ rob1,
                          const float* __restrict__ roW2, const float* __restrict__ rob2,
                          float* __restrict__ out, int n)
{
    __shared__ float mean[DIM];
    __shared__ float red[DIM];
    int c = threadIdx.x; // 128 threads
    mean[c] = csum[c] / (float)n;
    __syncthreads();
    float s = rob1[c];
#pragma unroll 8
    for (int k = 0; k < DIM; ++k) s += roW1[c * DIM + k] * mean[k];
    s = s > 0.0f ? s : 0.0f;          // relu(t)
    red[c] = s * roW2[c];
    __syncthreads();
    for (int o = 64; o > 0; o >>= 1) {
        if (c < o) red[c] += red[c + o];
        __syncthreads();
    }
    if (c == 0) out[0] = red[0] + rob2[0];
}

// edge importance: one wave per edge
__global__ void edge_imp_k(const float* __restrict__ hA, const float* __restrict__ hB,
                           const int* __restrict__ src, const int* __restrict__ dst,
                           const float* __restrict__ eib1, const float* __restrict__ eiW2,
                           const float* __restrict__ eib2,
                           const float* __restrict__ att0, const float* __restrict__ att1,
                           float* __restrict__ out, int E)
{
    int t = blockIdx.x * blockDim.x + threadIdx.x;
    int e = t >> 5;
    if (e >= E) return;
    int lane = t & 31;
    int c = lane * 4;
    int s = src[e], d = dst[e];
    float4 a  = *(const float4*)(hA + (size_t)s * DIM + c);
    float4 b  = *(const float4*)(hB + (size_t)d * DIM + c);
    float4 bb = *(const float4*)(eib1 + c);
    float4 w  = *(const float4*)(eiW2 + c);
    float p = 0.f, v;
    v = a.x + b.x + bb.x; v = v > 0.f ? v : 0.f; p += v * w.x;
    v = a.y + b.y + bb.y; v = v > 0.f ? v : 0.f; p += v * w.y;
    v = a.z + b.z + bb.z; v = v > 0.f ? v : 0.f; p += v * w.z;
    v = a.w + b.w + bb.w; v = v > 0.f ? v : 0.f; p += v * w.w;
#pragma unroll
    for (int o = 16; o > 0; o >>= 1) p += __shfl_xor(p, o, 32);
    if (lane == 0) {
        float im = 1.0f / (1.0f + expf(-(p + eib2[0])));
        out[e] = im * 0.5f * (att0[e] + att1[e]);
    }
}

__global__ void copy_k(const float* __restrict__ in, float* __restrict__ out, long long n)
{
    long long i = (long long)blockIdx.x * blockDim.x + threadIdx.x;
    if (i < n) out[i] = in[i];
}

extern "C" void kernel_launch(void* const* d_in, const int* in_sizes, int n_in,
                              void* d_out, int out_size, void* d_ws, size_t ws_size,
                              hipStream_t stream)
{
    const float* node_feats = (const float*)d_in[0];
    const int*   src        = (const int*)d_in[1];
    const int*   dst        = (const int*)d_in[2];
    const float* enc_W      = (const float*)d_in[3];
    const float* enc_b      = (const float*)d_in[4];
    const float* att_w      = (const float*)d_in[5];
    const float* att_b      = (const float*)d_in[6];
    const float* tr_W       = (const float*)d_in[7];
    const float* tr_b       = (const float*)d_in[8];
    const float* ro_W1      = (const float*)d_in[9];
    const float* ro_b1      = (const float*)d_in[10];
    const float* ro_W2      = (const float*)d_in[11];
    const float* ro_b2      = (const float*)d_in[12];
    const float* ei_W1      = (const float*)d_in[13];
    const float* ei_b1      = (const float*)d_in[14];
    const float* ei_W2      = (const float*)d_in[15];
    const float* ei_b2      = (const float*)d_in[16];

    const int N = NNODES, E = NEDGES, D = DIM;
    const long long ND = (long long)N * D;
    const int gemm_grid = (ROWTILES + 7) / 8;   // 313 blocks, 8 row tiles each

    float* ws = (float*)d_ws;
    size_t o = 0;
    float* h    = ws + o; o += (size_t)ND;
    float* agg  = ws + o; o += (size_t)ND;   // reused as hA after the layer loop
    float* hB   = ws + o; o += (size_t)ND;
    float* hs   = ws + o; o += N;
    float* hd   = ws + o; o += N;
    float* sc   = ws + o; o += E;
    float* att0 = ws + o; o += E;
    float* att1 = ws + o; o += E;
    float* csum = ws + o; o += 128;
    float* part = ws + o; o += 1024;
    float* redv = ws + o; o += 2;            // [0]=max, [1]=sum
    float* out  = (float*)d_out;

    // h = relu(node_feats @ enc_W.T + enc_b)
    gemm_wmma_f32<<<gemm_grid, 256, 0, stream>>>(node_feats, enc_W, D, 0, enc_b, nullptr, h, 1);

    for (int l = 0; l < 2; ++l) {
        float* attl = l ? att1 : att0;
        gemv2_kernel<<<(N + 255) / 256, 256, 0, stream>>>(h, att_w + l * 2 * D, hs, hd, N);
        edge_score_max<<<1024, 256, 0, stream>>>(hs, hd, src, dst, att_b, l, sc, part, E);
        reduce_max_k<<<1, 256, 0, stream>>>(part, redv, 1024);
        exp_sum_k<<<1024, 256, 0, stream>>>(sc, redv, part, E);
        reduce_sum_k<<<1, 256, 0, stream>>>(part, redv + 1, 1024);
        normalize_k<<<(E + 255) / 256, 256, 0, stream>>>(sc, redv + 1, attl, E);
        zero_f32<<<(int)((ND + 255) / 256), 256, 0, stream>>>(agg, ND);
        aggregate_k<<<(E * 32) / 256, 256, 0, stream>>>(h, src, dst, attl, agg, E);
        // h = relu(agg @ tr_W[l].T + tr_b[l]) + h   (in-place elementwise residual is safe)
        gemm_wmma_f32<<<gemm_grid, 256, 0, stream>>>(agg, tr_W + l * D * D, D, 0, tr_b + l * D, h, h, 1);
    }

    // graph score
    zero_f32<<<1, 128, 0, stream>>>(csum, 128);
    colsum_k<<<512, 128, 0, stream>>>(h, csum, N);
    readout_k<<<1, 128, 0, stream>>>(csum, ro_W1, ro_b1, ro_W2, ro_b2, out, N);

    // edge importance projections: hA = h @ ei_W1[:, :D].T ; hB = h @ ei_W1[:, D:].T
    float* hA = agg;
    gemm_wmma_f32<<<gemm_grid, 256, 0, stream>>>(h, ei_W1, 2 * D, 0, nullptr, nullptr, hA, 0);
    gemm_wmma_f32<<<gemm_grid, 256, 0, stream>>>(h, ei_W1, 2 * D, D, nullptr, nullptr, hB, 0);
    edge_imp_k<<<(E * 32) / 256, 256, 0, stream>>>(hA, hB, src, dst, ei_b1, ei_W2, ei_b2,
                                                   att0, att1, out + 1 + (size_t)ND, E);

    // h -> d_out[1:] (scalar copy; d_out+1 is only 4B aligned so no vector stores)
    copy_k<<<(int)((ND + 255) / 256), 256, 0, stream>>>(h, out + 1, ND);
}